// HybridTextureMLP_82695300317371
// MI455X (gfx1250) — compile-verified
//
#include <hip/hip_runtime.h>
#include <hip/hip_bf16.h>
#include <math.h>

// MI455X / gfx1250: wave32, WMMA f32 16x16x4 for the tiny fp32 MLP.
// Gather phase is lane-per-ray; MLP phase is 16-rays-per-WMMA-tile, 2 tiles/wave.

typedef __attribute__((ext_vector_type(2))) float v2f;
typedef __attribute__((ext_vector_type(8))) float v8f;

#define TS 2048
#define NCH 8
#define LDS_STRIDE 33          // 32 rays + 1 pad dword to avoid bank conflicts
#define WAVES_PER_BLOCK 8

__device__ __forceinline__ v8f wmma_f32_k4(v2f a, v2f b, v8f c) {
    // 8 args: (neg_a, A, neg_b, B, c_mod, C, reuse_a, reuse_b)
    return __builtin_amdgcn_wmma_f32_16x16x4_f32(false, a, false, b, (short)0, c,
                                                 false, false);
}

__global__ __launch_bounds__(256)
void hybrid_texture_mlp_kernel(const float* __restrict__ uv,
                               const float* __restrict__ viewdirs,
                               const float* __restrict__ texture,
                               const float* __restrict__ w1, const float* __restrict__ b1,
                               const float* __restrict__ w2, const float* __restrict__ b2,
                               const float* __restrict__ w3, const float* __restrict__ b3,
                               float* __restrict__ out, int nrays)
{
    __shared__ float smem[WAVES_PER_BLOCK * 16 * LDS_STRIDE];
    const int lane = threadIdx.x & 31;
    const int wid  = threadIdx.x >> 5;
    const int half = lane >> 4;      // 0: lanes 0-15, 1: lanes 16-31
    const int m    = lane & 15;
    float* ws = smem + wid * 16 * LDS_STRIDE;   // private 16x33 tile per wave

    const int waveBase = (blockIdx.x * WAVES_PER_BLOCK + wid) * 32;
    int ray = waveBase + lane;
    if (ray >= nrays) ray = nrays - 1;          // keep EXEC all-ones for WMMA

    // ---------------- gather: bilinear, align_corners, border clamp ----------
    float u = uv[2 * ray + 0], v = uv[2 * ray + 1];
    float fx = fminf(fmaxf((u + 1.0f) * (0.5f * (TS - 1)), 0.0f), (float)(TS - 1));
    float fy = fminf(fmaxf((v + 1.0f) * (0.5f * (TS - 1)), 0.0f), (float)(TS - 1));
    int x0 = (int)fx; if (x0 > TS - 2) x0 = TS - 2;   // keeps +1 texel in-bounds;
    int y0 = (int)fy; if (y0 > TS - 2) y0 = TS - 2;   // weight becomes exactly 1.0
    float wx = fx - (float)x0;
    float wy = fy - (float)y0;
    float w00 = (1.0f - wx) * (1.0f - wy), w01 = wx * (1.0f - wy);
    float w10 = (1.0f - wx) * wy,          w11 = wx * wy;
    int base00 = y0 * TS + x0;
    #pragma unroll
    for (int c = 0; c < NCH; ++c) {
        const float* p = texture + c * (TS * TS) + base00;
        float f00 = p[0], f01 = p[1], f10 = p[TS], f11 = p[TS + 1];
        ws[c * LDS_STRIDE + lane] = w00 * f00 + w01 * f01 + w10 * f10 + w11 * f11;
    }
    ws[ 8 * LDS_STRIDE + lane] = viewdirs[3 * ray + 0];
    ws[ 9 * LDS_STRIDE + lane] = viewdirs[3 * ray + 1];
    ws[10 * LDS_STRIDE + lane] = viewdirs[3 * ray + 2];
    ws[11 * LDS_STRIDE + lane] = 0.0f;           // K-padding row

    // ---------------- weights -> WMMA A layout (16x4 fp32, 2 VGPRs) ----------
    // lanes 0-15: K = k0, k0+1 ; lanes 16-31: K = k0+2, k0+3 (M = lane&15)
    v2f A1[3], A2[4], A3[4];
    #pragma unroll
    for (int kk = 0; kk < 3; ++kk) {
        int k0 = kk * 4 + half * 2;
        A1[kk].x = (k0     < 11) ? w1[m * 11 + k0    ] : 0.0f;
        A1[kk].y = (k0 + 1 < 11) ? w1[m * 11 + k0 + 1] : 0.0f;
    }
    #pragma unroll
    for (int kk = 0; kk < 4; ++kk) {
        int k0 = kk * 4 + half * 2;
        A2[kk].x = w2[m * 16 + k0];
        A2[kk].y = w2[m * 16 + k0 + 1];
        A3[kk].x = (m < 3) ? w3[m * 16 + k0    ] : 0.0f;  // W3 zero-padded 3->16 rows
        A3[kk].y = (m < 3) ? w3[m * 16 + k0 + 1] : 0.0f;
    }
    // biases folded into accumulator init (C/D: vgpr j -> row j+8*half, col = lane&15)
    v8f C1, C2, C3;
    #pragma unroll
    for (int j = 0; j < 8; ++j) {
        C1[j] = b1[j + 8 * half];
        C2[j] = b2[j + 8 * half];
        C3[j] = (half == 0 && j < 3) ? b3[j] : 0.0f;
    }

    // ---------------- two 16-ray tiles per wave ------------------------------
    #pragma unroll
    for (int t = 0; t < 2; ++t) {
        const int col = t * 16 + m;

        // layer 1: h1 = relu(W1 x + b1)   (3 K-chunks)
        v8f acc = C1;
        #pragma unroll
        for (int kk = 0; kk < 3; ++kk) {
            int k0 = kk * 4 + half * 2;
            v2f bm;
            bm.x = ws[(k0    ) * LDS_STRIDE + col];
            bm.y = ws[(k0 + 1) * LDS_STRIDE + col];
            acc = wmma_f32_k4(A1[kk], bm, acc);
        }
        #pragma unroll
        for (int j = 0; j < 8; ++j)
            ws[(j + 8 * half) * LDS_STRIDE + col] = fmaxf(acc[j], 0.0f);

        // layer 2: h2 = relu(W2 h1 + b2)  (4 K-chunks)
        v8f acc2 = C2;
        #pragma unroll
        for (int kk = 0; kk < 4; ++kk) {
            int k0 = kk * 4 + half * 2;
            v2f bm;
            bm.x = ws[(k0    ) * LDS_STRIDE + col];
            bm.y = ws[(k0 + 1) * LDS_STRIDE + col];
            acc2 = wmma_f32_k4(A2[kk], bm, acc2);
        }
        #pragma unroll
        for (int j = 0; j < 8; ++j)
            ws[(j + 8 * half) * LDS_STRIDE + col] = fmaxf(acc2[j], 0.0f);

        // layer 3: rgb = sigmoid(W3 h2 + b3)
        v8f acc3 = C3;
        #pragma unroll
        for (int kk = 0; kk < 4; ++kk) {
            int k0 = kk * 4 + half * 2;
            v2f bm;
            bm.x = ws[(k0    ) * LDS_STRIDE + col];
            bm.y = ws[(k0 + 1) * LDS_STRIDE + col];
            acc3 = wmma_f32_k4(A3[kk], bm, acc3);
        }
        // rows 0..2 live in vgprs 0..2 of lanes 0-15
        if (half == 0) {
            int r = waveBase + col;
            if (r < nrays) {
                out[3 * r + 0] = 1.0f / (1.0f + __expf(-acc3[0]));
                out[3 * r + 1] = 1.0f / (1.0f + __expf(-acc3[1]));
                out[3 * r + 2] = 1.0f / (1.0f + __expf(-acc3[2]));
            }
        }
    }
}

extern "C" void kernel_launch(void* const* d_in, const int* in_sizes, int n_in,
                              void* d_out, int out_size, void* d_ws, size_t ws_size,
                              hipStream_t stream) {
    const float* uv  = (const float*)d_in[0];
    const float* vd  = (const float*)d_in[1];
    const float* tex = (const float*)d_in[2];
    const float* w1  = (const float*)d_in[3];
    const float* b1  = (const float*)d_in[4];
    const float* w2  = (const float*)d_in[5];
    const float* b2  = (const float*)d_in[6];
    const float* w3  = (const float*)d_in[7];
    const float* b3  = (const float*)d_in[8];
    float* out = (float*)d_out;

    int nrays  = in_sizes[0] / 2;                       // uv is [B,2]
    int waves  = (nrays + 31) / 32;
    int blocks = (waves + WAVES_PER_BLOCK - 1) / WAVES_PER_BLOCK;
    hybrid_texture_mlp_kernel<<<blocks, 32 * WAVES_PER_BLOCK, 0, stream>>>(
        uv, vd, tex, w1, b1, w2, b2, w3, b3, out, nrays);
}